// LSTMClassifier_19885698580705
// MI455X (gfx1250) — compile-verified
//
#include <hip/hip_runtime.h>

// LSTM classifier: B=1024, T=1024, H=16, C=6
// One wave32 per 16-batch group. Transposed recurrence gates^T = W_hh @ h^T
// with (a) a K-permutation baked into the constant W_hh fragments so the
// D-tile register layout of h IS the B-fragment layout of h^T for the next
// step (register-to-register recurrence, no LDS/shuffles/barriers), and
// (b) the sigmoid half-scale folded into the i/f/o gate weight fragments so
// sigmoid is just v_tanh_f32 + fma. 20 chained V_WMMA_F32_16X16X4_F32/step.

typedef float v2f __attribute__((ext_vector_type(2)));
typedef float v8f __attribute__((ext_vector_type(8)));

#define B_TOT 1024
#define T_LEN 1024
#define H_DIM 16
#define N_CLS 6

#if __has_builtin(__builtin_amdgcn_tanhf)
#define HAVE_HW_TANH 1
__device__ __forceinline__ float hw_tanh(float x) {
    return __builtin_amdgcn_tanhf(x);                  // v_tanh_f32
}
// Gate pre-activations for i/f/o arrive PRE-SCALED by 0.5 (folded into A):
// sigmoid(x) = 0.5*tanh(x/2) + 0.5  ->  fma(0.5, tanh(y), 0.5), y = x/2.
__device__ __forceinline__ float sigmoid_prescaled(float y) {
    return fmaf(0.5f, __builtin_amdgcn_tanhf(y), 0.5f);
}
#else
#define HAVE_HW_TANH 0
__device__ __forceinline__ float hw_tanh(float x) {
    return fmaf(2.0f, __builtin_amdgcn_rcpf(1.0f + __expf(-2.0f * x)), -1.0f);
}
__device__ __forceinline__ float sigmoid_prescaled(float y) {   // y = x/2
    return __builtin_amdgcn_rcpf(1.0f + __expf(-2.0f * y));
}
#endif

// K-permutation: pi(k) = (k&2) ? 8 + 2*(k>>2) + (k&1) : 2*(k>>2) + (k&1).
// Applied to both A columns (constant W_hh load) and B rows (h state), it
// makes B-fragment slot (kf, v, half) want hidden (2kf + v + 8*half), which
// is exactly D/C-tile slot r = 2kf+v of the same lane.

__global__ __launch_bounds__(32) void lstm_wmma_kernel(
    const float* __restrict__ x,     // [B, T, 1]
    const float* __restrict__ W_ih,  // [64, 1]
    const float* __restrict__ W_hh,  // [64, 16]
    const float* __restrict__ b_ih,  // [64]
    const float* __restrict__ b_hh,  // [64]
    const float* __restrict__ W_fc,  // [6, 16]
    const float* __restrict__ b_fc,  // [6]
    float* __restrict__ out)         // [B, 6]
{
    const int lane  = threadIdx.x;   // 0..31 (wave32)
    const int l16   = lane & 15;
    const int half  = lane >> 4;     // 0 or 1
    const int bbase = blockIdx.x * 16;

    // ---- Constant A fragments of W_hh, K-permuted, sigmoid-scale folded ----
    // Tiles 0,1,3 (i,f,o gates) are scaled by 0.5; tile 2 (g gate) unscaled.
    // A[m][k] = s*W_hh[tile*16 + m][pi(k)]; slot (kf, v, half) holds
    // global k = 4kf + v + 2*half  ->  pi(k) = (half ? 8 : 0) + 2*kf + v.
    v2f a_hh[4][4];
    #pragma unroll
    for (int tile = 0; tile < 4; ++tile) {
        const float s = (tile == 2) ? 1.0f : 0.5f;
        #pragma unroll
        for (int kf = 0; kf < 4; ++kf) {
            const int row = (tile * 16 + l16) * H_DIM;
            const int j0  = (half ? 8 : 0) + 2 * kf;
            a_hh[tile][kf][0] = s * W_hh[row + j0 + 0];
            a_hh[tile][kf][1] = s * W_hh[row + j0 + 1];
        }
    }
    // Rank-1 x/bias K-slice: A cols {K0 = s*W_ih[gate], K1 = s*bias[gate]}.
    v2f a_ih[4];
    #pragma unroll
    for (int tile = 0; tile < 4; ++tile) {
        const float s = (tile == 2) ? 1.0f : 0.5f;
        const int g = tile * 16 + l16;
        a_ih[tile][0] = half ? 0.0f : s * W_ih[g];
        a_ih[tile][1] = half ? 0.0f : s * (b_ih[g] + b_hh[g]);
    }
    const float bx1 = half ? 0.0f : 1.0f;   // B row K1 = ones (bias multiplier)

    // State per lane (batch = l16 on both halves):
    // hpair[kf][v] = h[hidden 2kf+v+8*half]  (doubles as B fragment kf)
    // cst[r]       = c[hidden r+8*half]      (matches D slot r)
    v2f  hpair[4];
    float cst[8];
    #pragma unroll
    for (int kf = 0; kf < 4; ++kf) { hpair[kf][0] = 0.0f; hpair[kf][1] = 0.0f; }
    #pragma unroll
    for (int r = 0; r < 8; ++r) cst[r] = 0.0f;

    const float* xrow = x + (size_t)(bbase + l16) * T_LEN;

    #pragma unroll 1
    for (int t = 0; t < T_LEN; ++t) {
        const float xv = xrow[t];
        if (t + 64 < T_LEN) __builtin_prefetch(&xrow[t + 64], 0, 0);

        // B x-slice: row K0 = x_t[batch l16], row K1 = 1.0 (lanes 0-15 only).
        v2f b_x;
        b_x[0] = half ? 0.0f : xv;
        b_x[1] = bx1;

        // gates^T[tile] = x/bias slice + sum_kf W_hh_frag @ h^T_frag
        // h^T B fragments are the live hpair registers -- no data movement.
        v8f gacc[4];
        #pragma unroll
        for (int tile = 0; tile < 4; ++tile) {
            v8f c = {};
            c = __builtin_amdgcn_wmma_f32_16x16x4_f32(
                    false, a_ih[tile], false, b_x, (short)0, c, false, false);
            #pragma unroll
            for (int kf = 0; kf < 4; ++kf) {
                c = __builtin_amdgcn_wmma_f32_16x16x4_f32(
                        false, a_hh[tile][kf], false, hpair[kf],
                        (short)0, c, false, false);
            }
            gacc[tile] = c;
        }

        // Cell update. D slot r <-> (batch l16, hidden r+8*half) in every tile;
        // h slot r lives at hpair[r>>1][r&1] (same hidden index).
        // i/f/o pre-activations already halved -> sigmoid is tanh + fma.
        #pragma unroll
        for (int r = 0; r < 8; ++r) {
            const float ig = sigmoid_prescaled(gacc[0][r]);
            const float fg = sigmoid_prescaled(gacc[1][r]);
            const float gg = hw_tanh(gacc[2][r]);
            const float og = sigmoid_prescaled(gacc[3][r]);
            const float cc = fmaf(fg, cst[r], ig * gg);
            cst[r] = cc;
            hpair[r >> 1][r & 1] = og * hw_tanh(cc);
        }
    }

    // ---- Final FC: logits = h_last @ W_fc^T + b_fc ----
    // Lane holds hidden (r + 8*half) of batch l16: partial dot + half reduce.
    float logit[N_CLS];
    #pragma unroll
    for (int c = 0; c < N_CLS; ++c) {
        float acc = 0.0f;
        #pragma unroll
        for (int r = 0; r < 8; ++r)
            acc = fmaf(hpair[r >> 1][r & 1], W_fc[c * H_DIM + r + 8 * half], acc);
        logit[c] = acc + __shfl_xor(acc, 16, 32);
    }
    if (lane < 16) {
        #pragma unroll
        for (int c = 0; c < N_CLS; ++c)
            out[(size_t)(bbase + l16) * N_CLS + c] = logit[c] + b_fc[c];
    }
}

extern "C" void kernel_launch(void* const* d_in, const int* in_sizes, int n_in,
                              void* d_out, int out_size, void* d_ws, size_t ws_size,
                              hipStream_t stream) {
    const float* x    = (const float*)d_in[0];
    const float* W_ih = (const float*)d_in[1];
    const float* W_hh = (const float*)d_in[2];
    const float* b_ih = (const float*)d_in[3];
    const float* b_hh = (const float*)d_in[4];
    const float* W_fc = (const float*)d_in[5];
    const float* b_fc = (const float*)d_in[6];
    float* out = (float*)d_out;

    (void)in_sizes; (void)n_in; (void)out_size; (void)d_ws; (void)ws_size;

    dim3 grid(B_TOT / 16);   // 64 waves, one 16-batch group each
    dim3 block(32);          // one wave32
    lstm_wmma_kernel<<<grid, block, 0, stream>>>(x, W_ih, W_hh, b_ih, b_hh,
                                                 W_fc, b_fc, out);
}